// HeatmapBasisNFR_numJoints_Dynamic_MLP_align_80255758893472
// MI455X (gfx1250) — compile-verified
//
#include <hip/hip_runtime.h>

typedef __attribute__((ext_vector_type(2))) float v2f;
typedef __attribute__((ext_vector_type(8))) float v8f;

namespace {
constexpr int Bsz = 32768;
constexpr int Nj  = 17;
constexpr int Dd  = 2;
constexpr int Kk  = 3;
constexpr int H1  = 64;
constexpr int Hb  = 128;
constexpr int O1  = 3;
constexpr int Ob  = 12;

constexpr int WAVES = 4;          // waves per block (128 threads)
constexpr int HSTR  = Hb + 1;     // 129: padded LDS hidden row stride
constexpr int SSTR  = 20;         // stage row: 3 (k) + 3 (w) + 12 (params) + 2 pad

// Per-joint weight staging in LDS with biases folded in:
//  - W1 blocks: 4 K-rows x H; row 0,1 = W1, row 2 = b1 (ones-trick), row 3 = 0
//  - W2 blocks: (H+1) rows x 16 cols; rows 0..H-1 = W2 (cols >= O zeroed),
//    row H = b2 (cols >= O zeroed)
constexpr int OFF_KW1 = 0;                        // 4 x 64
constexpr int OFF_KW2 = OFF_KW1 + 4 * H1;         // 65 x 16
constexpr int OFF_WW1 = OFF_KW2 + (H1 + 1) * 16;  // 4 x 64
constexpr int OFF_WW2 = OFF_WW1 + 4 * H1;         // 65 x 16
constexpr int OFF_BW1 = OFF_WW2 + (H1 + 1) * 16;  // 4 x 128
constexpr int OFF_BW2 = OFF_BW1 + 4 * Hb;         // 129 x 16
constexpr int WTOT    = OFF_BW2 + (Hb + 1) * 16;  // 5168 floats

constexpr size_t OUTK = (size_t)Bsz * Nj * (Kk * 5); // start of k_js segment
constexpr size_t OUTM = OUTK + (size_t)Bsz * Nj;     // start of mask segment
}

__device__ __forceinline__ v8f wmma4(v2f a, v2f b, v8f c) {
  // D = A(16x4 f32) * B(4x16 f32) + C(16x16 f32)
  return __builtin_amdgcn_wmma_f32_16x16x4_f32(false, a, false, b,
                                               (short)0, c, false, false);
}

// Cooperative, single-pass staging of one joint's weights (padding zeros and
// folded biases written in the same pass -> one barrier after).
__device__ __forceinline__ void stage_joint_weights(
    float* __restrict__ lds, int joint, int tid, int nt,
    const float* __restrict__ kW1, const float* __restrict__ kb1,
    const float* __restrict__ kW2, const float* __restrict__ kb2,
    const float* __restrict__ wW1, const float* __restrict__ wb1,
    const float* __restrict__ wW2, const float* __restrict__ wb2,
    const float* __restrict__ bW1, const float* __restrict__ bb1,
    const float* __restrict__ bW2, const float* __restrict__ bb2)
{
  // W1 blocks (bias folded as K-row 2)
  for (int i = tid; i < 4 * H1; i += nt) {
    const int r = i / H1, c = i % H1;
    float kv = 0.0f, wv = 0.0f;
    if (r < Dd) {
      kv = kW1[(joint * Dd + r) * H1 + c];
      wv = wW1[(joint * Dd + r) * H1 + c];
    } else if (r == Dd) {
      kv = kb1[joint * H1 + c];
      wv = wb1[joint * H1 + c];
    }
    lds[OFF_KW1 + i] = kv;
    lds[OFF_WW1 + i] = wv;
  }
  for (int i = tid; i < 4 * Hb; i += nt) {
    const int r = i / Hb, c = i % Hb;
    float bv = 0.0f;
    if (r < Dd)       bv = bW1[(joint * Dd + r) * Hb + c];
    else if (r == Dd) bv = bb1[joint * Hb + c];
    lds[OFF_BW1 + i] = bv;
  }
  // W2 blocks (bias folded as row H; columns >= O zeroed)
  for (int i = tid; i < (H1 + 1) * 16; i += nt) {
    const int r = i >> 4, c = i & 15;
    float kv = 0.0f, wv = 0.0f;
    if (c < O1) {
      kv = (r < H1) ? kW2[(joint * H1 + r) * O1 + c] : kb2[joint * O1 + c];
      wv = (r < H1) ? wW2[(joint * H1 + r) * O1 + c] : wb2[joint * O1 + c];
    }
    lds[OFF_KW2 + i] = kv;
    lds[OFF_WW2 + i] = wv;
  }
  for (int i = tid; i < (Hb + 1) * 16; i += nt) {
    const int r = i >> 4, c = i & 15;
    float bv = 0.0f;
    if (c < Ob)
      bv = (r < Hb) ? bW2[(joint * Hb + r) * Ob + c] : bb2[joint * Ob + c];
    lds[OFF_BW2 + i] = bv;
  }
}

// One 16-point tile through a 2 -> H -> O per-joint MLP (ReLU between).
// Biases ride along as ones-rows, so every WMMA chain starts from inline-0 C.
// All weight reads are unconditional LDS loads; EXEC is all-1s at every WMMA.
template <int H, int O>
__device__ __forceinline__ void mlp16(
    v2f a, int lane,
    const float* __restrict__ lw1, const float* __restrict__ lw2,
    float* __restrict__ ldsH, float* __restrict__ stage, int stageOff)
{
  const int col  = lane & 15;
  const int half = lane >> 4;

  // ---- Layer 1: [x0 x1 1 0] @ [W1; b1; 0] -> ReLU -> LDS (16 x H, +1 pad)
  #pragma unroll
  for (int t = 0; t < H / 16; ++t) {
    const int c = t * 16 + col;
    v2f bf;                                   // B 4x16: v0={K0|K2}, v1={K1|K3}
    bf.x = lw1[(half * 2 + 0) * H + c];       // half1: row2 = b1
    bf.y = lw1[(half * 2 + 1) * H + c];       // half1: row3 = 0
    v8f acc = {};                             // -> inline-constant 0 SRC2
    acc = wmma4(a, bf, acc);
    #pragma unroll
    for (int r = 0; r < 8; ++r) {             // C: vgpr r, lane-half -> row
      const int m = r + half * 8;
      ldsH[m * HSTR + c] = fmaxf(acc[r], 0.0f);
    }
  }
  asm volatile("s_wait_dscnt 0" ::: "memory");

  // ---- Layer 2: [h 1] @ [W2; b2] -> stage (cols < O valid)
  v8f acc2 = {};
  #pragma unroll
  for (int kb = 0; kb < H; kb += 4) {
    const int kr = kb + half * 2;             // lane-halves split K 2+2
    v2f af, bf;
    af.x = ldsH[col * HSTR + kr + 0];         // A: row = this lane's point
    af.y = ldsH[col * HSTR + kr + 1];
    bf.x = lw2[(kr + 0) * 16 + col];          // padded cols -> zeros
    bf.y = lw2[(kr + 1) * 16 + col];
    acc2 = wmma4(af, bf, acc2);
  }
  {                                           // bias step: K-row H is all-ones
    const float b2c = lw2[H * 16 + col];      // staged b2 (padded)
    v2f af, bf;
    af.x = half ? 0.0f : 1.0f;                // K=H -> 1, K=H+1..H+3 -> 0
    af.y = 0.0f;
    bf.x = half ? 0.0f : b2c;                 // row H = b2, rows H+1..H+3 = 0
    bf.y = 0.0f;
    acc2 = wmma4(af, bf, acc2);
  }
  #pragma unroll
  for (int r = 0; r < 8; ++r) {
    const int m = r + half * 8;
    if (col < O) stage[m * SSTR + stageOff + col] = acc2[r];
  }
}

__global__ __launch_bounds__(WAVES * 32)
void heatmap_mdn_kernel(
    const float* __restrict__ pred,
    const float* __restrict__ kW1, const float* __restrict__ kb1,
    const float* __restrict__ kW2, const float* __restrict__ kb2,
    const float* __restrict__ wW1, const float* __restrict__ wb1,
    const float* __restrict__ wW2, const float* __restrict__ wb2,
    const float* __restrict__ bW1, const float* __restrict__ bb1,
    const float* __restrict__ bW2, const float* __restrict__ bb2,
    float* __restrict__ out)
{
  __shared__ float ldsW[WTOT];
  __shared__ float ldsH[WAVES][16 * HSTR];
  __shared__ float ldsS[WAVES][16 * SSTR];

  const int tid   = threadIdx.x;
  const int joint = blockIdx.y;               // all waves share this joint

  stage_joint_weights(ldsW, joint, tid, WAVES * 32,
                      kW1, kb1, kW2, kb2, wW1, wb1, wW2, wb2,
                      bW1, bb1, bW2, bb2);
  __syncthreads();

  const int wave = tid >> 5;
  const int lane = tid & 31;
  const int tile = blockIdx.x * WAVES + wave;

  const int col  = lane & 15;
  const int half = lane >> 4;

  // A fragment: [x0 x1 1 0] per point (ones-row K=2 carries the bias)
  const int b_idx = tile * 16 + col;          // valid address for all lanes
  const float* xp = pred + ((size_t)b_idx * Nj + joint) * Dd;
  const float x0 = xp[0];
  const float x1 = xp[1];
  v2f a;
  a.x = half ? 1.0f : x0;                     // K2 = 1 (bias row)
  a.y = half ? 0.0f : x1;                     // K3 = 0

  float* myH = ldsH[wave];
  float* myS = ldsS[wave];

  mlp16<H1, O1>(a, lane, ldsW + OFF_KW1, ldsW + OFF_KW2, myH, myS, 0); // k
  mlp16<H1, O1>(a, lane, ldsW + OFF_WW1, ldsW + OFF_WW2, myH, myS, 3); // w
  mlp16<Hb, Ob>(a, lane, ldsW + OFF_BW1, ldsW + OFF_BW2, myH, myS, 6); // params

  asm volatile("s_wait_dscnt 0" ::: "memory");

  // ---- scalar epilogue: lane p owns point p of the tile
  if (lane < 16) {
    const float* st = myS + lane * SSTR;

    // k_js = argmax(k_logits) + 1 (first max, like jnp.argmax)
    const float k0 = st[0], k1 = st[1], k2 = st[2];
    int kjs = 0; float kv = k0;
    if (k1 > kv) { kv = k1; kjs = 1; }
    if (k2 > kv) { kv = k2; kjs = 2; }
    kjs += 1;

    // softmax over w logits (max-subtracted)
    const float l0 = st[3], l1 = st[4], l2 = st[5];
    const float lm = fmaxf(l0, fmaxf(l1, l2));
    const float e0 = __expf(l0 - lm), e1 = __expf(l1 - lm), e2 = __expf(l2 - lm);
    const float inv = 1.0f / (e0 + e1 + e2);
    const float ws[3] = { e0 * inv, e1 * inv, e2 * inv };

    float mu[3][2], sg[3][2];
    #pragma unroll
    for (int i = 0; i < 3; ++i) {
      mu[i][0] = st[6 + i * 4 + 0];
      mu[i][1] = st[6 + i * 4 + 1];
      sg[i][0] = __expf(st[6 + i * 4 + 2]);
      sg[i][1] = __expf(st[6 + i * 4 + 3]);
    }

    // stable 3-element argsort of w descending (strict > keeps ties stable)
    float s0 = ws[0], s1 = ws[1], s2 = ws[2];
    int i0 = 0, i1 = 1, i2 = 2;
    if (s1 > s0) { float t = s0; s0 = s1; s1 = t; int ti = i0; i0 = i1; i1 = ti; }
    if (s2 > s1) { float t = s1; s1 = s2; s2 = t; int ti = i1; i1 = i2; i2 = ti; }
    if (s1 > s0) { float t = s0; s0 = s1; s1 = t; int ti = i0; i0 = i1; i1 = ti; }
    const int idx[3] = { i0, i1, i2 };

    const size_t pt = (size_t)b_idx * Nj + joint;
    float* op = out + pt * (Kk * 5);
    #pragma unroll
    for (int i = 0; i < 3; ++i) {
      op[i * 5 + 0] = ws[i];           // w stays in original order
      op[i * 5 + 1] = mu[idx[i]][0];   // mu/sigma sorted by w desc
      op[i * 5 + 2] = mu[idx[i]][1];
      op[i * 5 + 3] = sg[idx[i]][0];
      op[i * 5 + 4] = sg[idx[i]][1];
    }
    out[OUTK + pt] = (float)kjs;
    float* mp = out + OUTM + pt * Kk;
    #pragma unroll
    for (int i = 0; i < 3; ++i) mp[i] = (i < kjs) ? 1.0f : 0.0f;
  }
}

extern "C" void kernel_launch(void* const* d_in, const int* in_sizes, int n_in,
                              void* d_out, int out_size, void* d_ws, size_t ws_size,
                              hipStream_t stream) {
  (void)in_sizes; (void)n_in; (void)d_ws; (void)ws_size; (void)out_size;
  const float* pred = (const float*)d_in[0];
  const float* kW1  = (const float*)d_in[1];
  const float* kb1  = (const float*)d_in[2];
  const float* kW2  = (const float*)d_in[3];
  const float* kb2  = (const float*)d_in[4];
  const float* wW1  = (const float*)d_in[5];
  const float* wb1  = (const float*)d_in[6];
  const float* wW2  = (const float*)d_in[7];
  const float* wb2  = (const float*)d_in[8];
  const float* bW1  = (const float*)d_in[9];
  const float* bb1  = (const float*)d_in[10];
  const float* bW2  = (const float*)d_in[11];
  const float* bb2  = (const float*)d_in[12];
  float* out = (float*)d_out;

  dim3 grid(Bsz / 16 / WAVES, Nj);   // (512, 17): blockIdx.y = joint
  heatmap_mdn_kernel<<<grid, WAVES * 32, 0, stream>>>(
      pred, kW1, kb1, kW2, kb2, wW1, wb1, wW2, wb2, bW1, bb1, bW2, bb2, out);
}